// GumbelSampler_29772713296403
// MI455X (gfx1250) — compile-verified
//
#include <hip/hip_runtime.h>
#include <stdint.h>
#include <stddef.h>

#define V_DIM   50257
#define BLOCK   1024
#define CHUNK   4096                               // elements per array per chunk
#define EPB     (CHUNK / BLOCK)                    // 4 elements per thread per chunk
#define NBUF    3
#define NC      ((V_DIM + CHUNK - 1) / CHUNK)      // 13 chunks
#define EPS_F   1e-10f

// ---------------- CDNA5 async global->LDS copy path ----------------
#if __has_builtin(__builtin_amdgcn_global_load_async_to_lds_b32)
#define HAVE_ASYNC_LDS 1
#else
#define HAVE_ASYNC_LDS 0
#endif

typedef __attribute__((address_space(1))) int as1_int;
typedef __attribute__((address_space(3))) int as3_int;

__device__ __forceinline__ void async_copy_f32(const float* gsrc, float* ldst) {
#if HAVE_ASYNC_LDS
    __builtin_amdgcn_global_load_async_to_lds_b32(
        (as1_int*)(gsrc), (as3_int*)(ldst), 0, 0);
#else
    *ldst = *gsrc;   // synchronous fallback
#endif
}

template <int N>
__device__ __forceinline__ void wait_asynccnt() {
#if HAVE_ASYNC_LDS
#if __has_builtin(__builtin_amdgcn_s_wait_asynccnt)
    __builtin_amdgcn_s_wait_asynccnt(N);
#else
    asm volatile("s_wait_asynccnt %0" :: "n"(N) : "memory");
#endif
#endif
}

// ------------- monotone float <-> uint32 key mapping ---------------
__device__ __forceinline__ unsigned f2k(float f) {
    unsigned b = __float_as_uint(f);
    return b ^ (unsigned)(((int)b >> 31) | 0x80000000);
}
__device__ __forceinline__ float k2f(unsigned k) {
    unsigned m = (k >> 31) ? 0x80000000u : 0xFFFFFFFFu;
    return __uint_as_float(k ^ m);
}

// -------------------------------------------------------------------
// One block per row. The row (as monotone u32 keys, 201KB) lives in
// the 320KB WGP LDS. Async global->LDS staging (96KB, 3 buffers, 2
// chunks in flight => ~64KB/WGP outstanding) feeds the gumbel pass;
// the same 96KB is reused as 32 bank-rotated per-wave histograms for
// the radix select, then as reduction scratch. HBM traffic ~= the
// 2-read + 1-write floor (logits re-read is L2-hot, NT output stores
// avoid polluting L2).
// -------------------------------------------------------------------
__global__ __launch_bounds__(BLOCK)
void GumbelSampler_kernel(const float* __restrict__ logits,
                          const float* __restrict__ uin,
                          const int*   __restrict__ kptr,
                          float*       __restrict__ out)
{
    __shared__ unsigned gbuf[V_DIM];            // 201,028 B: row keys / masked logits
    __shared__ float    stage[NBUF][2][CHUNK];  //  98,304 B: async staging (reused)
    __shared__ unsigned hist[256];
    __shared__ unsigned s_prefix;
    __shared__ int      s_k;

    const int tid = threadIdx.x;
    const int row = blockIdx.x;
    const float* lrow = logits + (size_t)row * V_DIM;
    const float* urow = uin    + (size_t)row * V_DIM;
    float*       orow = out    + (size_t)row * V_DIM;

    // ---- Pass 1: async-stream logits+u (3-deep pipeline), gumbel keys -> gbuf ----
    {
        auto issue = [&](int c) {
            float* sbL = &stage[c % NBUF][0][0];
            float* sbU = &stage[c % NBUF][1][0];
            const int base = c * CHUNK;
#pragma unroll
            for (int e = 0; e < EPB; ++e) {
                int idx = base + e * BLOCK + tid;
                idx = idx < V_DIM ? idx : V_DIM - 1;   // clamp: uniform 8 issues/wave/chunk
                async_copy_f32(lrow + idx, sbL + e * BLOCK + tid);
                async_copy_f32(urow + idx, sbU + e * BLOCK + tid);
            }
        };
        issue(0);
        if (NC > 1) issue(1);
        for (int c = 0; c < NC; ++c) {
            if (c + 2 < NC) issue(c + 2);
            const int ahead = (NC - 1 - c) < 2 ? (NC - 1 - c) : 2;  // chunks newer than c
            if (ahead == 2)      wait_asynccnt<16>();
            else if (ahead == 1) wait_asynccnt<8>();
            else                 wait_asynccnt<0>();
            const float* sbL = &stage[c % NBUF][0][0];
            const float* sbU = &stage[c % NBUF][1][0];
#pragma unroll
            for (int e = 0; e < EPB; ++e) {
                const int idx = c * CHUNK + e * BLOCK + tid;
                if (idx < V_DIM) {
                    float l  = sbL[e * BLOCK + tid];
                    float uu = sbU[e * BLOCK + tid];
                    float noise = -__logf(-__logf(uu + EPS_F) + EPS_F);
                    gbuf[idx] = f2k(l + noise);   // store monotone key
                }
            }
        }
    }
    if (tid == 0) { s_prefix = 0u; s_k = *kptr; }
    __syncthreads();

    // ---- Exact k-th largest: 4-round radix select, per-wave privatized hists ----
    // whist[w][b] at word offset w*257+b  => bank (w+b)%64: hot bins of different
    // waves land in different banks (defeats the exponent-concentration hotspot).
    unsigned* whist = (unsigned*)&stage[0][0][0];      // 32*257 = 8224 words (~33KB)
    for (int shift = 24; shift >= 0; shift -= 8) {
        for (int j = tid; j < 32 * 257; j += BLOCK) whist[j] = 0u;
        __syncthreads();
        const unsigned prefix = s_prefix;
        const unsigned pmask  = (shift == 24) ? 0u : (0xFFFFFFFFu << (shift + 8));
        unsigned* myh = whist + (tid >> 5) * 257;
        for (int i = tid; i < V_DIM; i += BLOCK) {
            unsigned key = gbuf[i];
            if ((key & pmask) == prefix)
                atomicAdd(&myh[(key >> shift) & 255u], 1u);
        }
        __syncthreads();
        if (tid < 256) {
            unsigned s = 0;
#pragma unroll
            for (int w = 0; w < 32; ++w) s += whist[w * 257 + tid];
            hist[tid] = s;
        }
        __syncthreads();
        if (tid == 0) {
            int kk = s_k;
            unsigned cum = 0u;
            int b = 255;
            for (; b > 0; --b) {
                unsigned h = hist[b];
                if (cum + h >= (unsigned)kk) break;
                cum += h;
            }
            s_prefix = prefix | ((unsigned)b << shift);
            s_k = kk - (int)cum;
        }
        __syncthreads();
    }
    const float T = k2f(s_prefix);   // threshold = k-th largest gumbel logit

    // ---- Pass 2: masked logits (in place) + row max ----
    float* red = (float*)(whist + 32 * 257);   // 1024-float reduction scratch
    float lmax = -3.402823466e38f;
    for (int i = tid; i < V_DIM; i += BLOCK) {
        float l  = lrow[i];                        // 2nd logits read: L2-hot
        float g  = k2f(gbuf[i]);
        float sm = 1.0f / (1.0f + __expf(T - g));  // sigmoid(g - T)
        float m  = l * sm;
        gbuf[i]  = __float_as_uint(m);
        lmax     = fmaxf(lmax, m);
    }
    red[tid] = lmax;
    __syncthreads();
    for (int s = BLOCK / 2; s > 0; s >>= 1) {
        if (tid < s) red[tid] = fmaxf(red[tid], red[tid + s]);
        __syncthreads();
    }
    const float rmax = red[0];
    __syncthreads();

    // ---- Pass 3: sum of exp ----
    float lsum = 0.0f;
    for (int i = tid; i < V_DIM; i += BLOCK)
        lsum += __expf(__uint_as_float(gbuf[i]) - rmax);
    red[tid] = lsum;
    __syncthreads();
    for (int s = BLOCK / 2; s > 0; s >>= 1) {
        if (tid < s) red[tid] += red[tid + s];
        __syncthreads();
    }
    const float inv = 1.0f / red[0];

    // ---- Pass 4: softmax out, non-temporal coalesced stores ----
    for (int i = tid; i < V_DIM; i += BLOCK) {
        float v = __expf(__uint_as_float(gbuf[i]) - rmax) * inv;
        __builtin_nontemporal_store(v, orow + i);
    }
}

extern "C" void kernel_launch(void* const* d_in, const int* in_sizes, int n_in,
                              void* d_out, int out_size, void* d_ws, size_t ws_size,
                              hipStream_t stream) {
    const float* logits = (const float*)d_in[0];
    const float* u      = (const float*)d_in[1];
    const int*   kp     = (const int*)d_in[2];
    float*       out    = (float*)d_out;
    (void)d_ws; (void)ws_size; (void)n_in; (void)out_size;

    const int B = in_sizes[0] / V_DIM;   // 2048
    GumbelSampler_kernel<<<dim3(B), dim3(BLOCK), 0, stream>>>(logits, u, kp, out);
}